// SimpleAttentionLayer_94489281074
// MI455X (gfx1250) — compile-verified
//
#include <hip/hip_runtime.h>
#include <hip/hip_bf16.h>

#define N_NODES 50000
#define D_FEAT  128
#define N_EDGES 800000
#define LN_EPS  1e-5f

typedef __attribute__((ext_vector_type(2))) float v2f;
typedef __attribute__((ext_vector_type(8))) float v8f;

// ---------------------------------------------------------------------------
// 0) Workspace init: zero agg + denom, seed segment-max keys with key(-inf).
// ---------------------------------------------------------------------------
__global__ __launch_bounds__(256) void init_ws_kernel(float* __restrict__ agg,
                                                      float* __restrict__ denom,
                                                      unsigned* __restrict__ segmax_u) {
    int i = blockIdx.x * blockDim.x + threadIdx.x;
    if (i < N_NODES * D_FEAT) agg[i] = 0.0f;
    if (i < N_NODES) {
        denom[i] = 0.0f;
        segmax_u[i] = 0x007fffffu;  // monotone key of -inf
    }
}

// ---------------------------------------------------------------------------
// 1) h = x @ W_lin^T via native fp32 WMMA (V_WMMA_F32_16X16X4_F32).
//    One wave -> one 16x16 output tile. 8 waves/block -> 16x128 strip.
//    A (16x4): lanes 0-15 hold K={0,1}, lanes 16-31 hold K={2,3}.
//    B (4x16): B[k][n] = W_lin[n][k]; lanes 0-15 K={0,1}, lanes 16-31 K={2,3},
//              N = lane%16.  C/D: VGPR r -> row r (+8 for upper half), col=lane%16.
// ---------------------------------------------------------------------------
__global__ __launch_bounds__(256) void gemm_wmma_kernel(const float* __restrict__ x,
                                                        const float* __restrict__ W,
                                                        float* __restrict__ h) {
    const int lane = threadIdx.x & 31;
    const int wv   = threadIdx.x >> 5;   // 0..7 : column tile
    const int m    = lane & 15;
    const int hi   = lane >> 4;          // 0 or 1 (half-wave)

    const long row = (long)blockIdx.x * 16 + m;        // A row
    const int  col = wv * 16 + m;                      // B column (output feature)

    const float* __restrict__ xr = x + row * D_FEAT + 2 * hi;
    const float* __restrict__ wr = W + (long)col * D_FEAT + 2 * hi;

    v8f c = {};
#pragma unroll
    for (int k = 0; k < D_FEAT; k += 4) {
        v2f a; a.x = xr[k]; a.y = xr[k + 1];
        v2f b; b.x = wr[k]; b.y = wr[k + 1];
        c = __builtin_amdgcn_wmma_f32_16x16x4_f32(false, a, false, b,
                                                  (short)0, c, false, false);
    }

    const long rbase = (long)blockIdx.x * 16;
#pragma unroll
    for (int r = 0; r < 8; ++r) {
        h[(rbase + r + 8 * hi) * D_FEAT + wv * 16 + m] = c[r];
    }
}

// ---------------------------------------------------------------------------
// 2) s_src[n] = h[n,:].a_src ; s_dst[n] = h[n,:].a_dst  (one wave per node)
// ---------------------------------------------------------------------------
__global__ __launch_bounds__(256) void node_scores_kernel(const float* __restrict__ h,
                                                          const float* __restrict__ Wattn,
                                                          float* __restrict__ s_src,
                                                          float* __restrict__ s_dst) {
    const int lane = threadIdx.x & 31;
    const int node = blockIdx.x * 8 + (threadIdx.x >> 5);
    if (node >= N_NODES) return;

    const float4 hv = ((const float4*)(h + (long)node * D_FEAT))[lane];
    const float4 as = ((const float4*)(Wattn))[lane];
    const float4 ad = ((const float4*)(Wattn + D_FEAT))[lane];

    float ss = hv.x * as.x + hv.y * as.y + hv.z * as.z + hv.w * as.w;
    float sd = hv.x * ad.x + hv.y * ad.y + hv.z * ad.z + hv.w * ad.w;
#pragma unroll
    for (int off = 16; off > 0; off >>= 1) {
        ss += __shfl_down(ss, off);
        sd += __shfl_down(sd, off);
    }
    if (lane == 0) { s_src[node] = ss; s_dst[node] = sd; }
}

// ---------------------------------------------------------------------------
// 3) Edge scores + leaky ReLU + segment max (monotone-key u32 atomicMax).
// ---------------------------------------------------------------------------
__global__ __launch_bounds__(256) void edge_scores_kernel(const int* __restrict__ src,
                                                          const int* __restrict__ dst,
                                                          const float* __restrict__ s_src,
                                                          const float* __restrict__ s_dst,
                                                          float* __restrict__ scores,
                                                          unsigned* __restrict__ segmax_u) {
    int e = blockIdx.x * blockDim.x + threadIdx.x;
    if (e >= N_EDGES) return;
    float sc = s_src[src[e]] + s_dst[dst[e]];
    sc = (sc >= 0.0f) ? sc : 0.2f * sc;
    scores[e] = sc;
    int bits = __float_as_int(sc);
    unsigned key = (unsigned)(bits ^ ((bits >> 31) | 0x80000000));
    atomicMax(&segmax_u[dst[e]], key);
}

// ---------------------------------------------------------------------------
// 4) Decode segment max; nodes with no in-edges (still -inf) -> 0.
// ---------------------------------------------------------------------------
__global__ __launch_bounds__(256) void node_prep_kernel(const unsigned* __restrict__ segmax_u,
                                                        float* __restrict__ segmax_f) {
    int n = blockIdx.x * blockDim.x + threadIdx.x;
    if (n >= N_NODES) return;
    unsigned key = segmax_u[n];
    int bits = (key & 0x80000000u) ? (int)(key ^ 0x80000000u) : ~(int)key;
    float m = __int_as_float(bits);
    segmax_f[n] = isfinite(m) ? m : 0.0f;
}

// ---------------------------------------------------------------------------
// 5) ex = exp(score - segmax[dst]) ; denom[dst] += ex
// ---------------------------------------------------------------------------
__global__ __launch_bounds__(256) void edge_exp_kernel(const int* __restrict__ dst,
                                                       const float* __restrict__ scores,
                                                       const float* __restrict__ segmax_f,
                                                       float* __restrict__ ex,
                                                       float* __restrict__ denom) {
    int e = blockIdx.x * blockDim.x + threadIdx.x;
    if (e >= N_EDGES) return;
    int d = dst[e];
    float v = __expf(scores[e] - segmax_f[d]);
    ex[e] = v;
    atomicAdd(&denom[d], v);
}

// ---------------------------------------------------------------------------
// 6) agg[dst] += h[src] * alpha  (one wave per edge: coalesced 512B gather,
//    4 coalesced global_atomic_add_f32 per lane)
// ---------------------------------------------------------------------------
__global__ __launch_bounds__(256) void edge_aggregate_kernel(const int* __restrict__ src,
                                                             const int* __restrict__ dst,
                                                             const float* __restrict__ h,
                                                             const float* __restrict__ ex,
                                                             const float* __restrict__ denom,
                                                             float* __restrict__ agg) {
    const int lane = threadIdx.x & 31;
    const long e = (long)blockIdx.x * 8 + (threadIdx.x >> 5);
    if (e >= N_EDGES) return;

    const int s = src[e];
    const int d = dst[e];
    const float den = denom[d];
    const float alpha = ex[e] / ((den > 0.0f) ? den : 1.0f);

    const float* __restrict__ hs = h + (long)s * D_FEAT;
    float* __restrict__ ad = agg + (long)d * D_FEAT;
#pragma unroll
    for (int t = 0; t < 4; ++t) {
        int idx = lane + 32 * t;
        atomicAdd(&ad[idx], hs[idx] * alpha);
    }
}

// ---------------------------------------------------------------------------
// 7) out = LayerNorm(agg + x) * scale + bias  (one wave per node)
// ---------------------------------------------------------------------------
__global__ __launch_bounds__(256) void layernorm_kernel(const float* __restrict__ agg,
                                                        const float* __restrict__ x,
                                                        const float* __restrict__ scale,
                                                        const float* __restrict__ bias,
                                                        float* __restrict__ out) {
    const int lane = threadIdx.x & 31;
    const int n = blockIdx.x * 8 + (threadIdx.x >> 5);
    if (n >= N_NODES) return;

    const float4 a  = ((const float4*)(agg + (long)n * D_FEAT))[lane];
    const float4 xv = ((const float4*)(x + (long)n * D_FEAT))[lane];
    float v0 = a.x + xv.x, v1 = a.y + xv.y, v2 = a.z + xv.z, v3 = a.w + xv.w;

    float sum = v0 + v1 + v2 + v3;
#pragma unroll
    for (int off = 16; off > 0; off >>= 1) sum += __shfl_down(sum, off);
    sum = __shfl(sum, 0);
    const float mu = sum * (1.0f / 128.0f);

    float d0 = v0 - mu, d1 = v1 - mu, d2 = v2 - mu, d3 = v3 - mu;
    float vs = d0 * d0 + d1 * d1 + d2 * d2 + d3 * d3;
#pragma unroll
    for (int off = 16; off > 0; off >>= 1) vs += __shfl_down(vs, off);
    vs = __shfl(vs, 0);
    const float inv = rsqrtf(vs * (1.0f / 128.0f) + LN_EPS);

    const float4 sc = ((const float4*)scale)[lane];
    const float4 bi = ((const float4*)bias)[lane];
    float4 o;
    o.x = d0 * inv * sc.x + bi.x;
    o.y = d1 * inv * sc.y + bi.y;
    o.z = d2 * inv * sc.z + bi.z;
    o.w = d3 * inv * sc.w + bi.w;
    ((float4*)(out + (long)n * D_FEAT))[lane] = o;
}

// ---------------------------------------------------------------------------
// Launch
// ---------------------------------------------------------------------------
extern "C" void kernel_launch(void* const* d_in, const int* in_sizes, int n_in,
                              void* d_out, int out_size, void* d_ws, size_t ws_size,
                              hipStream_t stream) {
    const float* x       = (const float*)d_in[0];   // N x 128
    const float* W_lin   = (const float*)d_in[1];   // 128 x 128
    const float* W_attn  = (const float*)d_in[2];   // 1 x 256
    const float* ln_scale= (const float*)d_in[3];   // 128
    const float* ln_bias = (const float*)d_in[4];   // 128
    const int*   ei      = (const int*)d_in[5];     // 2 x E
    const int*   src     = ei;
    const int*   dst     = ei + N_EDGES;
    float*       out     = (float*)d_out;

    // Workspace layout (floats)
    float* ws        = (float*)d_ws;
    float* h         = ws;                                       // N*D
    float* agg       = h + (size_t)N_NODES * D_FEAT;             // N*D
    float* s_src     = agg + (size_t)N_NODES * D_FEAT;           // N
    float* s_dst     = s_src + N_NODES;                          // N
    unsigned* segmax_u = (unsigned*)(s_dst + N_NODES);           // N
    float* segmax_f  = (float*)(segmax_u + N_NODES);             // N
    float* denom     = segmax_f + N_NODES;                       // N
    float* scores    = denom + N_NODES;                          // E
    float* ex        = scores + N_EDGES;                         // E

    const int B = 256;

    init_ws_kernel<<<(N_NODES * D_FEAT + B - 1) / B, B, 0, stream>>>(agg, denom, segmax_u);

    gemm_wmma_kernel<<<N_NODES / 16, B, 0, stream>>>(x, W_lin, h);            // 3125 blocks

    node_scores_kernel<<<(N_NODES + 7) / 8, B, 0, stream>>>(h, W_attn, s_src, s_dst);

    edge_scores_kernel<<<(N_EDGES + B - 1) / B, B, 0, stream>>>(src, dst, s_src, s_dst,
                                                                scores, segmax_u);

    node_prep_kernel<<<(N_NODES + B - 1) / B, B, 0, stream>>>(segmax_u, segmax_f);

    edge_exp_kernel<<<(N_EDGES + B - 1) / B, B, 0, stream>>>(dst, scores, segmax_f, ex, denom);

    edge_aggregate_kernel<<<(N_EDGES + 7) / 8, B, 0, stream>>>(src, dst, h, ex, denom, agg);

    layernorm_kernel<<<(N_NODES + 7) / 8, B, 0, stream>>>(agg, x, ln_scale, ln_bias, out);
}